// ResDGFLayer_14448269984557
// MI455X (gfx1250) — compile-verified
//
#include <hip/hip_runtime.h>
#include <hip/hip_bf16.h>

typedef float v2f __attribute__((ext_vector_type(2)));
typedef float v8f __attribute__((ext_vector_type(8)));

#define B_SZ    4
#define N_TOK   4096
#define D_DIM   128
#define MT      64      // row/col tile
#define XS      132     // padded LDS stride for 128-wide x tiles (4*r % 64 distinct -> no bank conflicts)
#define AS      68      // padded LDS stride for 64-wide adj tile
#define LN_EPS  1e-5f
#define SIG_EPS 1e-6f

__device__ __forceinline__ v2f ld2(const float* p) { return *(const v2f*)p; }

// D = A(16x4 f32) * B(4x16 f32) + C(16x16 f32), wave32
__device__ __forceinline__ v8f wmma4(v2f a, v2f b, v8f c) {
  return __builtin_amdgcn_wmma_f32_16x16x4_f32(
      /*neg_a=*/false, a, /*neg_b=*/false, b,
      /*c_mod=*/(short)0, c, /*reuse_a=*/false, /*reuse_b=*/false);
}

// Async copy of a 64x128 f32 tile (row-major, ld=128) from global into LDS at
// byte offset lds_base (row stride XS floats). 2048 B128 chunks / 256 threads
// = 8 per thread. Tracked by ASYNCcnt; pair with s_wait_asynccnt + barrier.
__device__ __forceinline__ void async_copy_tile(unsigned lds_base,
                                                const float* __restrict__ src,
                                                int tid) {
#pragma unroll
  for (int i = 0; i < 8; ++i) {
    const int idx = tid + i * 256;
    const int r = idx >> 5, c4 = (idx & 31) << 2;
    const float* g = src + r * D_DIM + c4;
    const unsigned l = lds_base + (unsigned)((r * XS + c4) * 4);
    asm volatile("global_load_async_to_lds_b128 %0, %1, off"
                 :
                 : "v"(l), "v"(g)
                 : "memory");
  }
}

__device__ __forceinline__ void wait_async_all() {
  asm volatile("s_wait_asynccnt 0x0" ::: "memory");
}

// One m-tile: Gram via WMMA -> clamped multi-head RBF -> diffusion WMMA accum.
// All pointers are compile-time-fixed at each call site => pure ds_ ops.
__device__ __forceinline__ void process_tile(
    const float* xr, const float* xcc, float* adj,
    const float* x2r, const float* x2cc, const float (&rd)[4],
    v8f (&oacc)[4], int tm, int tn0, int dg0, int aL, int kh,
    int mrow, int ncol) {
  // ---- Gram: G = Xr * Xc^T  (two 16x16 subtiles per wave, K = 128)
  v8f g0 = (v8f){0, 0, 0, 0, 0, 0, 0, 0};
  v8f g1 = (v8f){0, 0, 0, 0, 0, 0, 0, 0};
  const float* ar  = xr + (tm * 16 + aL) * XS + kh;
  const float* br0 = xcc + (tn0 * 16 + aL) * XS + kh;
  const float* br1 = xcc + ((tn0 + 1) * 16 + aL) * XS + kh;
#pragma unroll 8
  for (int kk = 0; kk < D_DIM / 4; ++kk) {
    v2f av  = ld2(ar + kk * 4);
    v2f bv0 = ld2(br0 + kk * 4);
    v2f bv1 = ld2(br1 + kk * 4);
    g0 = wmma4(av, bv0, g0);
    g1 = wmma4(av, bv1, g1);
  }

  // ---- elementwise: dist -> mean-over-heads RBF, stage adj tile in LDS
#pragma unroll
  for (int g = 0; g < 8; ++g) {
    float x2rv = x2r[mrow + g];
    float d0 = fmaxf(x2rv + x2cc[ncol] - 2.0f * g0[g], 0.0f);
    float d1 = fmaxf(x2rv + x2cc[ncol + 16] - 2.0f * g1[g], 0.0f);
    float a0 = 0.25f * (__expf(-d0 * rd[0]) + __expf(-d0 * rd[1]) +
                        __expf(-d0 * rd[2]) + __expf(-d0 * rd[3]));
    float a1 = 0.25f * (__expf(-d1 * rd[0]) + __expf(-d1 * rd[1]) +
                        __expf(-d1 * rd[2]) + __expf(-d1 * rd[3]));
    adj[(mrow + g) * AS + ncol]      = a0;
    adj[(mrow + g) * AS + ncol + 16] = a1;
  }
  __syncthreads();  // adj tile visible to all waves

  // ---- diffusion: out[64x128] += adj[64x64] * Xc[64x128]  (K = 64)
  const float* aar = adj + (tm * 16 + aL) * AS + kh;
#pragma unroll 4
  for (int kk = 0; kk < MT / 4; ++kk) {
    v2f av = ld2(aar + kk * 4);
    const int krow = kk * 4 + kh;
    const float* bb = xcc + krow * XS;
#pragma unroll
    for (int j = 0; j < 4; ++j) {
      const int c = (dg0 + j) * 16 + aL;
      v2f bv;
      bv.x = bb[c];
      bv.y = bb[XS + c];
      oacc[j] = wmma4(av, bv, oacc[j]);
    }
  }
}

// ---------------------------------------------------------------------------
// Kernel 1: squared row norms  x2[b*N+n] = sum_d x[b,n,d]^2
// ---------------------------------------------------------------------------
__global__ void x2_kernel(const float* __restrict__ x, float* __restrict__ x2) {
  int row = blockIdx.x * blockDim.x + threadIdx.x;   // 0 .. B*N-1
  const float4* xr = (const float4*)(x + (size_t)row * D_DIM);
  float s = 0.0f;
#pragma unroll
  for (int i = 0; i < D_DIM / 4; ++i) {
    float4 v = xr[i];
    s += v.x * v.x + v.y * v.y + v.z * v.z + v.w * v.w;
  }
  x2[row] = s;
}

// ---------------------------------------------------------------------------
// Kernel 2: fused  dist -> RBF(mean over heads) -> diffusion -> proj -> ELU
//           -> residual -> LayerNorm          (one 64-row tile per workgroup)
// Double-buffered column tiles via GLOBAL_LOAD_ASYNC_TO_LDS (ASYNCcnt),
// two-phase unrolled so every LDS pointer is statically known.
// ---------------------------------------------------------------------------
__global__ __launch_bounds__(256, 1) void resdgf_main(
    const float* __restrict__ x, const float* __restrict__ log_sigmas,
    const float* __restrict__ W, const float* __restrict__ pbias,
    const float* __restrict__ gamma, const float* __restrict__ beta,
    const float* __restrict__ x2g, float* __restrict__ out) {
  extern __shared__ float smem[];
  float* xr    = smem;                // 64 x 132  (row tile, persistent; residual source)
  float* xc0   = xr + MT * XS;        // 64 x 132  (col tile, buffer 0; reused as staging)
  float* xc1   = xc0 + MT * XS;       // 64 x 132  (col tile, buffer 1)
  float* adj   = xc1 + MT * XS;       // 64 x 68   (adjacency tile)
  float* x2r   = adj + MT * AS;       // 64
  float* x2c0  = x2r + MT;            // 64
  float* x2c1  = x2c0 + MT;           // 64
  float* red   = x2c1 + MT;           // 512 (partial sums / sumsq)
  float* stats = red + MT * 8;        // 128 (mu, rsig)

  const int tid  = threadIdx.x;
  const int lane = tid & 31;
  const int wave = tid >> 5;
  const int b    = blockIdx.y;
  const int row0 = blockIdx.x * MT;

  // LDS byte offsets (dynamic LDS starts at offset 0; no static __shared__)
  const unsigned xr_off  = (unsigned)((xr - smem) * 4);
  const unsigned xc_off0 = (unsigned)((xc0 - smem) * 4);
  const unsigned xc_off1 = (unsigned)((xc1 - smem) * 4);

  // per-head reciprocal denominators: 1 / (2*sigma^2 + eps)
  float rd[4];
#pragma unroll
  for (int h = 0; h < 4; ++h) {
    float s = __expf(log_sigmas[h]);
    rd[h] = 1.0f / (2.0f * s * s + SIG_EPS);
  }

  // ---- prologue: async-stage row tile + first column tile
  const float* xbatch = x + (size_t)b * N_TOK * D_DIM;
  async_copy_tile(xr_off, xbatch + (size_t)row0 * D_DIM, tid);
  async_copy_tile(xc_off0, xbatch, tid);  // tile mt=0 (col0 = 0)
  if (tid < MT) {
    x2r[tid]  = x2g[b * N_TOK + row0 + tid];
    x2c0[tid] = x2g[b * N_TOK + tid];
  }

  // wave -> subtile ownership
  const int tm   = wave & 3;        // 16-row subtile of the 64-row tile
  const int half = wave >> 2;       // 0/1
  const int tn0  = half * 2;        // gram: col subtiles {tn0, tn0+1}
  const int dg0  = half * 4;        // diffusion/proj: d subtiles dg0..dg0+3
  const int aL   = lane & 15;
  const int kh   = (lane >> 4) * 2; // K sub-offset per lane half (f32 WMMA operand layout)

  v8f oacc[4];
#pragma unroll
  for (int j = 0; j < 4; ++j) oacc[j] = (v8f){0, 0, 0, 0, 0, 0, 0, 0};

  const int mrow = tm * 16 + (lane >> 4) * 8;  // C-layout row base for this lane
  const int ncol = tn0 * 16 + aL;              // C-layout col for gram subtile 0

  // ====================== m-tile loop (2 tiles / iter) =====================
  for (int mt = 0; mt < N_TOK / MT; mt += 2) {
    // -------- phase A: consume buffer 0 (tile mt), prefetch into buffer 1
    wait_async_all();   // my fills of xc0 have landed
    __syncthreads();    // everyone's fills landed; readers of xc1 are done
    {
      const int nc = (mt + 1) * MT;  // mt+1 < 64 always (64 tiles, even count)
      async_copy_tile(xc_off1, xbatch + (size_t)nc * D_DIM, tid);
      if (tid < MT) x2c1[tid] = x2g[b * N_TOK + nc + tid];
    }
    process_tile(xr, xc0, adj, x2r, x2c0, rd, oacc,
                 tm, tn0, dg0, aL, kh, mrow, ncol);

    // -------- phase B: consume buffer 1 (tile mt+1), prefetch into buffer 0
    wait_async_all();
    __syncthreads();
    if (mt + 2 < N_TOK / MT) {
      const int nc = (mt + 2) * MT;
      async_copy_tile(xc_off0, xbatch + (size_t)nc * D_DIM, tid);
      if (tid < MT) x2c0[tid] = x2g[b * N_TOK + nc + tid];
    }
    process_tile(xr, xc1, adj, x2r, x2c1, rd, oacc,
                 tm, tn0, dg0, aL, kh, mrow, ncol);
  }
  // =========================================================================

  __syncthreads();  // all diffusion reads done -> reuse xc0 as staging
  float* pbuf = xc0;
#pragma unroll
  for (int j = 0; j < 4; ++j)
#pragma unroll
    for (int g = 0; g < 8; ++g)
      pbuf[(mrow + g) * XS + (dg0 + j) * 16 + aL] = oacc[j][g];
  __syncthreads();

  // ---- projection: P = out_tile @ W^T   (B[k][n] = W[n][k], streamed via L2)
  v8f pacc[4];
#pragma unroll
  for (int j = 0; j < 4; ++j) pacc[j] = (v8f){0, 0, 0, 0, 0, 0, 0, 0};
  const float* par = pbuf + (tm * 16 + aL) * XS + kh;
#pragma unroll 4
  for (int kk = 0; kk < D_DIM / 4; ++kk) {
    v2f av = ld2(par + kk * 4);
    const int kg = kk * 4 + kh;
#pragma unroll
    for (int j = 0; j < 4; ++j) {
      const int n = (dg0 + j) * 16 + aL;       // output feature index
      v2f bv = ld2(W + (size_t)n * D_DIM + kg);
      pacc[j] = wmma4(av, bv, pacc[j]);
    }
  }
  __syncthreads();  // all reads of pbuf done before overwriting with y

  // ---- bias + ELU + residual -> y staged back into pbuf
#pragma unroll
  for (int j = 0; j < 4; ++j)
#pragma unroll
    for (int g = 0; g < 8; ++g) {
      const int r = mrow + g;
      const int c = (dg0 + j) * 16 + aL;
      float v = pacc[j][g] + pbias[c];
      float e = v > 0.0f ? v : (__expf(v) - 1.0f);   // ELU
      pbuf[r * XS + c] = e + xr[r * XS + c];         // residual
    }
  __syncthreads();

  // ---- LayerNorm stats: 4 partial reducers per row
  {
    const int r = tid & 63, part = tid >> 6;
    const float* yr = pbuf + r * XS + part * 32;
    float s = 0.0f, sq = 0.0f;
#pragma unroll
    for (int i = 0; i < 32; ++i) {
      float v = yr[i];
      s += v; sq += v * v;
    }
    red[part * 64 + r]       = s;
    red[256 + part * 64 + r] = sq;
  }
  __syncthreads();
  if (tid < MT) {
    float s  = red[tid] + red[64 + tid] + red[128 + tid] + red[192 + tid];
    float sq = red[256 + tid] + red[320 + tid] + red[384 + tid] + red[448 + tid];
    float mu  = s * (1.0f / 128.0f);
    float var = sq * (1.0f / 128.0f) - mu * mu;
    stats[tid]      = mu;
    stats[64 + tid] = rsqrtf(var + LN_EPS);
  }
  __syncthreads();

  // ---- normalize + affine, coalesced store
  float* orow = out + ((size_t)b * N_TOK + row0) * D_DIM;
  for (int i = tid; i < MT * D_DIM; i += 256) {
    const int r = i >> 7, c = i & 127;
    float v = (pbuf[r * XS + c] - stats[r]) * stats[64 + r];
    orow[i] = v * gamma[c] + beta[c];
  }
}

// ---------------------------------------------------------------------------
extern "C" void kernel_launch(void* const* d_in, const int* in_sizes, int n_in,
                              void* d_out, int out_size, void* d_ws, size_t ws_size,
                              hipStream_t stream) {
  const float* x     = (const float*)d_in[0];
  const float* ls    = (const float*)d_in[1];
  const float* W     = (const float*)d_in[2];
  const float* pb    = (const float*)d_in[3];
  const float* gamma = (const float*)d_in[4];
  const float* beta  = (const float*)d_in[5];
  float* out = (float*)d_out;
  float* x2  = (float*)d_ws;  // B*N floats = 64 KB scratch

  x2_kernel<<<(B_SZ * N_TOK) / 256, 256, 0, stream>>>(x, x2);

  dim3 grid(N_TOK / MT, B_SZ);
  size_t smem = (size_t)(MT * XS * 3 + MT * AS + MT * 3 + MT * 8 + 128) * sizeof(float);
  resdgf_main<<<grid, 256, smem, stream>>>(x, ls, W, pb, gamma, beta, x2, out);
}